// GATCQNetwork_89653147337561
// MI455X (gfx1250) — compile-verified
//
#include <hip/hip_runtime.h>
#include <math.h>

// ---------------------------------------------------------------------------
// GATCQNetwork on MI455X (gfx1250, wave32)
//   N=256 nodes, F=255 feats (padded to 256), H=4096, OUT=32640, E=65280
// GAT layers -> dense 256x256x256 fp32 GEMMs via v_wmma_f32_16x16x4_f32.
// Final layers are HBM-bandwidth-bound GEMVs (1.07GB + 535MB of weights).
// ---------------------------------------------------------------------------

#define NN   256
#define FF   255
#define HH   4096
#define OUTN 32640
#define EE   65280
#define K4   65280   // N*F
#define NEG_SLOPE 0.2f

typedef float v2f __attribute__((ext_vector_type(2)));
typedef float v8f __attribute__((ext_vector_type(8)));

// ---------------- pack kernels (zero-pad to 256x256) -----------------------

__global__ void pack_x_kernel(const float* __restrict__ x, float* __restrict__ xp) {
    int i = blockIdx.x * 256 + threadIdx.x;          // 0..65535
    int f = i & 255;
    int n = i >> 8;
    xp[i] = (f < FF) ? x[n * FF + f] : 0.0f;
}

// Wp[k][n] = W[n][k]  (B operand layout, K x N row-major), zero padded
__global__ void pack_Wt_kernel(const float* __restrict__ W, float* __restrict__ Wp) {
    int i = blockIdx.x * 256 + threadIdx.x;          // 0..65535
    int n = i & 255;
    int k = i >> 8;
    Wp[i] = (k < FF && n < FF) ? W[n * FF + k] : 0.0f;
}

// ---------------- edge multiplicity counts (graph fixed across layers) -----

__global__ void count_edges_kernel(const int* __restrict__ ei, int* __restrict__ C) {
    int e = blockIdx.x * 256 + threadIdx.x;          // 0..EE-1
    if (e < EE) {
        int s = ei[e];
        int d = ei[EE + e];
        atomicAdd(&C[d * NN + s], 1);
    }
}

// ---------------- fp32 WMMA GEMM: D[MxN] = A[MxK] * B[KxN] (+bias) ---------
// 256x256x256, one wave per 16x16 tile, 4 waves per block, 64 blocks.
// v_wmma_f32_16x16x4_f32: A 16x4 (2 VGPR), B 4x16 (2 VGPR), C/D 16x16 (8 VGPR).

__global__ __launch_bounds__(128) void gemm256_wmma_kernel(
        const float* __restrict__ A, const float* __restrict__ B,
        float* __restrict__ D, const float* __restrict__ bias) {
    const int wave = threadIdx.x >> 5;               // 0..3
    const int tile = blockIdx.x * 4 + wave;          // 0..255
    const int tm   = (tile >> 4) << 4;               // tile row base
    const int tn   = (tile & 15) << 4;               // tile col base
    const int lane = threadIdx.x & 31;
    const int half = lane >> 4;                      // 0: lanes 0-15, 1: 16-31
    const int l15  = lane & 15;

    v8f acc = {};
    for (int k0 = 0; k0 < NN; k0 += 4) {
        // A fragment: lanes 0-15 hold K = k0+0,k0+1 ; lanes 16-31 K = k0+2,k0+3
        const float* ap = A + (tm + l15) * NN + (k0 + 2 * half);
        v2f a; a.x = ap[0]; a.y = ap[1];
        // B fragment: mirrored K-pair split across lane halves, N striped on lanes
        const float* bp = B + (k0 + 2 * half) * NN + (tn + l15);
        v2f b; b.x = bp[0]; b.y = bp[NN];
        acc = __builtin_amdgcn_wmma_f32_16x16x4_f32(
                  false, a, false, b, (short)0, acc, false, false);
    }
    // C/D layout: VGPR r -> M = r + 8*half, N = l15
    const int col = tn + l15;
    const float badd = (bias && col < FF) ? bias[col] : 0.0f;
    #pragma unroll
    for (int r = 0; r < 8; ++r) {
        int row = tm + r + 8 * half;
        D[row * NN + col] = acc[r] + badd;
    }
}

// ---------------- per-node attention logits --------------------------------
// alpha_src[n] = h[n,:255] . a_src ; alpha_dst[n] = h[n,:255] . a_dst

__global__ void node_alphas_kernel(const float* __restrict__ h,
                                   const float* __restrict__ a_src,
                                   const float* __restrict__ a_dst,
                                   float* __restrict__ alpha_src,
                                   float* __restrict__ alpha_dst) {
    __shared__ float s1[256];
    __shared__ float s2[256];
    const int n = blockIdx.x, t = threadIdx.x;
    float hv = h[n * NN + t];
    float as = (t < FF) ? a_src[t] : 0.0f;
    float ad = (t < FF) ? a_dst[t] : 0.0f;
    s1[t] = hv * as;
    s2[t] = hv * ad;
    __syncthreads();
    for (int off = 128; off > 0; off >>= 1) {
        if (t < off) { s1[t] += s1[t + off]; s2[t] += s2[t + off]; }
        __syncthreads();
    }
    if (t == 0) { alpha_src[n] = s1[0]; alpha_dst[n] = s2[0]; }
}

// ---------------- dense attention rows (segment softmax over dst) ----------
// A[dst][src] = cnt * exp(e - m) / sum ;  e = leaky_relu(as[src] + ad[dst])

__global__ void attn_rows_kernel(const int* __restrict__ C,
                                 const float* __restrict__ alpha_src,
                                 const float* __restrict__ alpha_dst,
                                 float* __restrict__ Aout) {
    __shared__ float red[256];
    const int dst = blockIdx.x, src = threadIdx.x;
    int c = C[dst * NN + src] + (src == dst ? 1 : 0);   // self-loop
    float e = alpha_src[src] + alpha_dst[dst];
    e = (e > 0.0f) ? e : NEG_SLOPE * e;
    red[src] = (c > 0) ? e : -3.0e38f;
    __syncthreads();
    for (int off = 128; off > 0; off >>= 1) {
        if (src < off) red[src] = fmaxf(red[src], red[src + off]);
        __syncthreads();
    }
    float m = red[0];
    __syncthreads();
    float ex = (c > 0) ? (float)c * __expf(e - m) : 0.0f;
    red[src] = ex;
    __syncthreads();
    for (int off = 128; off > 0; off >>= 1) {
        if (src < off) red[src] += red[src + off];
        __syncthreads();
    }
    float s = red[0];
    Aout[dst * NN + src] = ex / s;
}

// ---------------- relu + flatten (drop pad column) -------------------------

__global__ void relu_flatten_kernel(const float* __restrict__ g, float* __restrict__ y1) {
    int i = blockIdx.x * 256 + threadIdx.x;          // 0..K4-1
    if (i < K4) {
        int n = i / FF, f = i - n * FF;
        float v = g[n * NN + f];
        y1[i] = (v > 0.0f) ? v : 0.0f;
    }
}

// ---------------- y2 = relu(W4 @ y1 + b4) : 4096 x 65280, BW-bound ---------
// one 256-thread block per row, float4 streaming loads, LDS tree reduce

__global__ __launch_bounds__(256) void gemv_w4_kernel(
        const float* __restrict__ W4, const float* __restrict__ y1,
        const float* __restrict__ b4, float* __restrict__ y2) {
    __shared__ float red[256];
    const int r = blockIdx.x, t = threadIdx.x;
    const float4* wrow = (const float4*)(W4 + (size_t)r * K4);
    const float4* yv   = (const float4*)y1;
    float acc = 0.0f;
    for (int i = t; i < K4 / 4; i += 256) {
        float4 w = wrow[i];
        float4 y = yv[i];
        acc += w.x * y.x + w.y * y.y + w.z * y.z + w.w * y.w;
    }
    red[t] = acc;
    __syncthreads();
    for (int off = 128; off > 0; off >>= 1) {
        if (t < off) red[t] += red[t + off];
        __syncthreads();
    }
    if (t == 0) {
        float v = red[0] + b4[r];
        y2[r] = (v > 0.0f) ? v : 0.0f;
    }
}

// ---------------- out = Wo @ y2 + bo : 32640 x 4096, BW-bound --------------
// one wave32 per row, float4 loads, shuffle reduce

__global__ __launch_bounds__(256) void gemv_out_kernel(
        const float* __restrict__ Wo, const float* __restrict__ y2,
        const float* __restrict__ bo, float* __restrict__ out) {
    const int row  = (blockIdx.x * blockDim.x + threadIdx.x) >> 5;
    const int lane = threadIdx.x & 31;
    if (row >= OUTN) return;
    const float4* wrow = (const float4*)(Wo + (size_t)row * HH);
    const float4* yv   = (const float4*)y2;
    float acc = 0.0f;
    for (int i = lane; i < HH / 4; i += 32) {
        float4 w = wrow[i];
        float4 y = yv[i];
        acc += w.x * y.x + w.y * y.y + w.z * y.z + w.w * y.w;
    }
    for (int off = 16; off > 0; off >>= 1)
        acc += __shfl_down(acc, off, 32);
    if (lane == 0) out[row] = acc + bo[row];
}

// ---------------------------------------------------------------------------

extern "C" void kernel_launch(void* const* d_in, const int* in_sizes, int n_in,
                              void* d_out, int out_size, void* d_ws, size_t ws_size,
                              hipStream_t stream) {
    const float* x  = (const float*)d_in[0];
    const int*   ei = (const int*)  d_in[1];
    const float* W[3]  = { (const float*)d_in[2], (const float*)d_in[6],  (const float*)d_in[10] };
    const float* As[3] = { (const float*)d_in[3], (const float*)d_in[7],  (const float*)d_in[11] };
    const float* Ad[3] = { (const float*)d_in[4], (const float*)d_in[8],  (const float*)d_in[12] };
    const float* Bi[3] = { (const float*)d_in[5], (const float*)d_in[9],  (const float*)d_in[13] };
    const float* W4 = (const float*)d_in[14];
    const float* b4 = (const float*)d_in[15];
    const float* Wo = (const float*)d_in[16];
    const float* bo = (const float*)d_in[17];
    float* out = (float*)d_out;

    // workspace layout (floats)
    float* ws   = (float*)d_ws;
    float* buf0 = ws;                 // 65536  padded node features (ping)
    float* buf1 = buf0 + NN * NN;     // 65536  padded node features (pong)
    float* hbuf = buf1 + NN * NN;     // 65536  h = x @ W^T
    float* Wp   = hbuf + NN * NN;     // 65536  packed W^T
    float* Aatt = Wp   + NN * NN;     // 65536  dense attention matrix
    int*   Cnt  = (int*)(Aatt + NN * NN); // 65536 ints: edge multiplicities
    float* asc  = (float*)(Cnt + NN * NN);// 256
    float* adc  = asc + NN;               // 256
    float* y1   = adc + NN;               // 65280
    float* y2   = y1 + K4;                // 4096

    // ---- graph counts (fixed across the 3 layers) ----
    hipMemsetAsync(Cnt, 0, NN * NN * sizeof(int), stream);
    count_edges_kernel<<<EE / 256, 256, 0, stream>>>(ei, Cnt);

    // ---- pad input features ----
    pack_x_kernel<<<256, 256, 0, stream>>>(x, buf0);

    float* cur = buf0;
    float* nxt = buf1;
    for (int l = 0; l < 3; ++l) {
        pack_Wt_kernel<<<256, 256, 0, stream>>>(W[l], Wp);
        // h = cur @ W^T                                  (WMMA fp32)
        gemm256_wmma_kernel<<<64, 128, 0, stream>>>(cur, Wp, hbuf, nullptr);
        node_alphas_kernel<<<256, 256, 0, stream>>>(hbuf, As[l], Ad[l], asc, adc);
        attn_rows_kernel<<<256, 256, 0, stream>>>(Cnt, asc, adc, Aatt);
        // next = Aatt @ h + b                            (WMMA fp32)
        gemm256_wmma_kernel<<<64, 128, 0, stream>>>(Aatt, hbuf, nxt, Bi[l]);
        float* t = cur; cur = nxt; nxt = t;
    }

    relu_flatten_kernel<<<(K4 + 255) / 256, 256, 0, stream>>>(cur, y1);
    gemv_w4_kernel<<<HH, 256, 0, stream>>>(W4, y1, b4, y2);
    gemv_out_kernel<<<(OUTN * 32 + 255) / 256, 256, 0, stream>>>(Wo, y2, bo, out);
}